// AnomalyGenerator_33904471835476
// MI455X (gfx1250) — compile-verified
//
#include <hip/hip_runtime.h>
#include <hip/hip_bf16.h>

typedef __attribute__((ext_vector_type(2))) float v2f;
typedef __attribute__((ext_vector_type(4))) float v4f;
typedef __attribute__((ext_vector_type(8))) float v8f;

#define GRID 33
#define HO   56
#define WO   56
#define NPIX (HO*WO)      // 3136
#define NPIX4 (NPIX/4)    // 784
#define CCH  768
#define NB2  32

#if __has_builtin(__builtin_amdgcn_wmma_f32_16x16x4_f32)
#define HAVE_WMMA_F32 1
#else
#define HAVE_WMMA_F32 0
#endif

// CDNA5 async global->LDS copy (ASYNCcnt-tracked). Inline asm: the clang
// builtin's parameter types use CUDA language address spaces HIP source can't
// spell. LDS operand = low 32 bits of the flat shared-aperture address
// (== LDS byte offset per the aperture mapping).
__device__ __forceinline__ void async_load_b128(const void* gsrc, void* ldst) {
    unsigned lds_off = (unsigned)(unsigned long long)ldst;
    asm volatile("global_load_async_to_lds_b128 %0, %1, off"
                 :
                 : "v"(lds_off), "v"(gsrc)
                 : "memory");
}

__device__ __forceinline__ void wait_asynccnt0() {
#if __has_builtin(__builtin_amdgcn_s_wait_asynccnt)
    __builtin_amdgcn_s_wait_asynccnt(0);
#else
    asm volatile("s_wait_asynccnt 0" ::: "memory");
#endif
}

__device__ __forceinline__ float fade_f(float t) {
    return t * t * t * (t * (t * 6.0f - 15.0f) + 10.0f);
}

// ---------------------------------------------------------------------------
// Kernel A: per-image (32 blocks x 128 threads)
//   Perlin 64x64 -> WMMA f32 bilinear resize (separable, weights in LDS) ->
//   threshold -> noise_mask / mask_out_b / lab_out.
// ---------------------------------------------------------------------------
__global__ __launch_bounds__(128)
void anomaly_mask_kernel(const float* __restrict__ mask,
                         const float* __restrict__ labels,
                         const float* __restrict__ angles,
                         const float* __restrict__ chance,
                         const int*   __restrict__ scale_x,
                         const int*   __restrict__ scale_y,
                         float* __restrict__ nm_out,   // ws: 32*3136 noise_mask
                         int*   __restrict__ mask_out, // 32*3136
                         int*   __restrict__ lab_out)  // 32
{
    __shared__ float sGc[GRID * GRID];
    __shared__ float sGs[GRID * GRID];
    __shared__ float sPN[64 * 64];   // Perlin 64x64; later reused for T2
    __shared__ float sT1[64 * 64];   // Wy * PN
    __shared__ float sW [64 * 64];   // bilinear weights W[y][k]; rows >=56 zero
    __shared__ int   sAny;

    const int b2  = blockIdx.x;
    const int tid = threadIdx.x;
    if (tid == 0) sAny = 0;

    // ---- phase 0: gradient table (cos/sin of angles) + zero weights ----
    const float* ang = angles + b2 * (GRID * GRID);
    for (int p = tid; p < GRID * GRID; p += 128) {
        float s, c;
        __sincosf(ang[p], &s, &c);
        sGc[p] = c;
        sGs[p] = s;
    }
    for (int p = tid; p < 64 * 64; p += 128)
        sW[p] = 0.0f;
    __syncthreads();

    // ---- phase 1: Perlin 64x64 (all threads) + weight fill (tid<56) ----
    const int sx = 1 << scale_x[b2];
    const int sy = 1 << scale_y[b2];
    for (int p = tid; p < 64 * 64; p += 128) {
        int   i  = p >> 6, j = p & 63;
        int   ix = i * sx, jy = j * sy;
        int   c0 = ix >> 6, c1 = jy >> 6;
        float fu = (float)(ix & 63) * (1.0f / 64.0f);
        float fv = (float)(jy & 63) * (1.0f / 64.0f);
        int   g  = c0 * GRID + c1;
        float n00 = fu * sGc[g] + fv * sGs[g];
        float n10 = (fu - 1.0f) * sGc[g + GRID] + fv * sGs[g + GRID];
        float n01 = fu * sGc[g + 1] + (fv - 1.0f) * sGs[g + 1];
        float n11 = (fu - 1.0f) * sGc[g + GRID + 1] + (fv - 1.0f) * sGs[g + GRID + 1];
        float tu = fade_f(fu), tv = fade_f(fv);
        float l0 = n00 + tu * (n10 - n00);
        float l1 = n01 + tu * (n11 - n01);
        sPN[p] = 1.41421356237f * (l0 + tv * (l1 - l0));
    }
    if (tid < HO) {
        // source coord s in [0.071, 62.93]: no clipping, i0+1 <= 63 always
        float s  = ((float)tid + 0.5f) * (64.0f / 56.0f) - 0.5f;
        int   i0 = (int)floorf(s);
        float w  = s - (float)i0;
        sW[tid * 64 + i0]     = 1.0f - w;
        sW[tid * 64 + i0 + 1] = w;
    }
    __syncthreads();

#if HAVE_WMMA_F32
    // ---- phase 2: separable bilinear resize as two 64x64x64 f32 WMMAs ----
    // Lane layout (ISA 7.12.2, wave32):
    //   A:  A(m = lane&15, k = v + 2*(lane>>4)) per 4-wide K step
    //   B:  B(k = v + 2*(lane>>4), n = lane&15)  (K split mirrors A)
    //   C/D: D(m = r + 8*(lane>>4), n = lane&15), r = 0..7
    // Tile loop index forced scalar (readfirstlane) so loop control is
    // s_cbranch and EXEC stays all-1s through every v_wmma (ISA 7.12 req).
    const int wid  = __builtin_amdgcn_readfirstlane(tid >> 5);
    const int lane = tid & 31;
    const int lm   = lane & 15;
    const int lh   = lane >> 4;

    // pass 1: T1 = W * PN
    for (int t = wid; t < 16; t += 4) {
        const int mt = t >> 2, nt = t & 3;
        const int arow = (mt * 16 + lm) * 64;
        const int bcol = nt * 16 + lm;
        v8f acc = {};
#pragma unroll
        for (int kt = 0; kt < 16; ++kt) {
            const int k0 = kt * 4 + 2 * lh;
            v2f a, b;
            a.x = sW[arow + k0];
            a.y = sW[arow + k0 + 1];
            b.x = sPN[(k0    ) * 64 + bcol];
            b.y = sPN[(k0 + 1) * 64 + bcol];
            acc = __builtin_amdgcn_wmma_f32_16x16x4_f32(
                false, a, false, b, (short)0, acc, false, false);
        }
#pragma unroll
        for (int r = 0; r < 8; ++r)
            sT1[(mt * 16 + r + 8 * lh) * 64 + bcol] = acc[r];
    }
    __syncthreads();

    // pass 2: T2 = T1 * W^T; B(k,n) = W[n][k] (contiguous pair -> b64 load)
    for (int t = wid; t < 16; t += 4) {
        const int mt = t >> 2, nt = t & 3;
        const int arow = (mt * 16 + lm) * 64;
        const int bcol = nt * 16 + lm;
        v8f acc = {};
#pragma unroll
        for (int kt = 0; kt < 16; ++kt) {
            const int k0 = kt * 4 + 2 * lh;
            v2f a, b;
            a.x = sT1[arow + k0];
            a.y = sT1[arow + k0 + 1];
            b.x = sW[bcol * 64 + k0];
            b.y = sW[bcol * 64 + k0 + 1];
            acc = __builtin_amdgcn_wmma_f32_16x16x4_f32(
                false, a, false, b, (short)0, acc, false, false);
        }
#pragma unroll
        for (int r = 0; r < 8; ++r)
            sPN[(mt * 16 + r + 8 * lh) * 64 + bcol] = acc[r];
    }
    __syncthreads();
    const float* sT2 = sPN;
#else
    // Fallback: direct scalar bilinear into sT1 (stride-64 layout)
    for (int p = tid; p < NPIX; p += 128) {
        int   y  = p / WO, x = p - y * WO;
        float syf = ((float)y + 0.5f) * (64.0f / 56.0f) - 0.5f;
        float sxf = ((float)x + 0.5f) * (64.0f / 56.0f) - 0.5f;
        int   y0 = (int)floorf(syf), x0 = (int)floorf(sxf);
        float wy = syf - (float)y0, wx = sxf - (float)x0;
        float v00 = sPN[y0 * 64 + x0],       v01 = sPN[y0 * 64 + x0 + 1];
        float v10 = sPN[(y0 + 1) * 64 + x0], v11 = sPN[(y0 + 1) * 64 + x0 + 1];
        float top = v00 + wx * (v01 - v00);
        float bot = v10 + wx * (v11 - v10);
        sT1[y * 64 + x] = top + wy * (bot - top);
    }
    __syncthreads();
    const float* sT2 = sT1;
#endif

    // ---- phase 3: threshold + mask logic + outputs ----
    const float  lab = labels[b2 & 15];
    const float  chc = chance[b2];
    const float* mkp = mask + (size_t)(b2 & 15) * NPIX;
    int any = 0;
    for (int p = tid; p < NPIX; p += 128) {
        int   y   = p / WO, x = p - y * WO;
        float pnv = sT2[y * 64 + x];
        float thr = (pnv > 0.5f) ? 1.0f : 0.0f;
        if (chc > 0.5f) thr = 0.0f;
        float m  = mkp[p];
        float nm = (1.0f - lab) * (1.0f - m) * thr;
        nm_out[(size_t)b2 * NPIX + p] = nm;
        int mob = ((m + nm) > 0.0f) ? 1 : 0;
        mask_out[(size_t)b2 * NPIX + p] = mob;
        any |= mob;
    }
    if (any) atomicOr(&sAny, 1);
    __syncthreads();
    if (tid == 0) lab_out[b2] = (lab > 0.0f || sAny) ? 1 : 0;
}

// ---------------------------------------------------------------------------
// Kernel B: bandwidth-bound streaming perturbation.
//   Grid (32, 24), 256 threads; each block owns one image x 32 channels,
//   stages the 12.5 KB noise_mask into LDS via async B128 loads, then
//   streams input(RT: fits L2, read 2x) / noise(NT) -> out(NT) as float4.
// ---------------------------------------------------------------------------
__global__ __launch_bounds__(256)
void perturb_kernel(const float* __restrict__ input,
                    const float* __restrict__ noise,
                    const float* __restrict__ nmask,
                    float* __restrict__ out)
{
    __shared__ __align__(16) float sM[NPIX];

    const int b2    = blockIdx.x;
    const int cbase = blockIdx.y * 32;
    const int tid   = threadIdx.x;

    const v4f* gnm = (const v4f*)(nmask + (size_t)b2 * NPIX);
    for (int p = tid; p < NPIX4; p += 256)
        async_load_b128(gnm + p, (v4f*)sM + p);
    wait_asynccnt0();
    __syncthreads();

    const v4f* inb = (const v4f*)input + (size_t)((b2 & 15) * CCH + cbase) * NPIX4;
    const v4f* nzb = (const v4f*)noise + (size_t)(b2 * CCH + cbase) * NPIX4;
    v4f*       ob  = (v4f*)out         + (size_t)(b2 * CCH + cbase) * NPIX4;
    const v4f* sM4 = (const v4f*)sM;

    for (int c = 0; c < 32; ++c) {
        const v4f* ic = inb + c * NPIX4;
        const v4f* nc = nzb + c * NPIX4;
        v4f*       oc = ob  + c * NPIX4;
        if (c + 1 < 32)  // warm next channel of the streamed operand
            __builtin_prefetch(nc + NPIX4 + tid, 0, 0);
        for (int p = tid; p < NPIX4; p += 256) {
            v4f a = ic[p];
            v4f n = __builtin_nontemporal_load(nc + p);
            v4f m = sM4[p];
            __builtin_nontemporal_store(a + n * m, oc + p);
        }
    }
}

// ---------------------------------------------------------------------------
extern "C" void kernel_launch(void* const* d_in, const int* in_sizes, int n_in,
                              void* d_out, int out_size, void* d_ws, size_t ws_size,
                              hipStream_t stream)
{
    const float* input  = (const float*)d_in[0];  // 16*768*56*56
    const float* mask   = (const float*)d_in[1];  // 16*1*56*56
    const float* labels = (const float*)d_in[2];  // 16
    const float* noise  = (const float*)d_in[3];  // 32*768*56*56
    const float* angles = (const float*)d_in[4];  // 32*33*33
    const float* chance = (const float*)d_in[5];  // 32
    const int*   sx     = (const int*)d_in[6];    // 32
    const int*   sy     = (const int*)d_in[7];    // 32

    float* out = (float*)d_out;
    const size_t PER = (size_t)NB2 * CCH * NPIX;  // 77,070,336 perturbed floats
    int* mout = (int*)(out + PER);                // 32*3136 int32
    int* lout = mout + (size_t)NB2 * NPIX;        // 32 int32
    float* nm = (float*)d_ws;                     // 32*3136 floats scratch

    anomaly_mask_kernel<<<dim3(NB2), dim3(128), 0, stream>>>(
        mask, labels, angles, chance, sx, sy, nm, mout, lout);
    perturb_kernel<<<dim3(NB2, CCH / 32), dim3(256), 0, stream>>>(
        input, noise, nm, out);
}